// SymmetryDistanceLoss_72962904425128
// MI455X (gfx1250) — compile-verified
//
#include <hip/hip_runtime.h>
#include <stdint.h>

// ---------------------------------------------------------------------------
// SymmetryDistanceLoss on MI455X (gfx1250)
//
// sym = A*p + t for 6 symmetry ops == an 18x4 matrix times homogeneous
// points. Pipeline:
//   - TDM (tensor_load_to_lds / TENSORcnt) double-buffers 1024-point tiles
//     of the HBM point stream into LDS.
//   - V_WMMA_F32_16X16X4_F32 applies the stacked 18x4 transform to 16-point
//     groups (two WMMAs: rows 0-15 and rows 16-17).
//   - 3 cross-half shuffles redistribute D so each lane owns 3 full (x,y,z)
//     triples -> voxel index -> L2-resident gather from `closest` -> distance.
//   - wave32 shuffle reduction + global_atomic_add_f32.
// ---------------------------------------------------------------------------

typedef unsigned int u32;
typedef u32   __attribute__((ext_vector_type(4))) u32x4;
typedef int   __attribute__((ext_vector_type(4))) i32x4;
typedef int   __attribute__((ext_vector_type(8))) i32x8;
typedef float __attribute__((ext_vector_type(2))) v2f;
typedef float __attribute__((ext_vector_type(8))) v8f;

#define TILE_PTS 1024          // points per TDM tile
#define TILES_PER_BLOCK 4      // tiles per workgroup (N assumed divisible)
#define BLOCK_THREADS 256      // 8 wave32
#define PTS_PER_WAVE (TILE_PTS / (BLOCK_THREADS / 32))   // 128

// --- TDM: DMA nFloats f32 from gptr into LDS at byte offset lds_off -------
// D# per CDNA5 ISA ch.7/8: 1-D tensor, data_size=4B, tile_dim0=tensor_dim0.
__device__ __forceinline__ void tdm_load_tile(const float* gptr, u32 lds_off,
                                              u32 nFloats) {
  unsigned long long ga = (unsigned long long)(uintptr_t)gptr;
  u32x4 g0;
  g0[0] = 1u;                                   // count=1, user mode
  g0[1] = lds_off;                              // lds_addr (bytes)
  g0[2] = (u32)(ga & 0xFFFFFFFFu);              // global_addr[31:0]
  g0[3] = (u32)((ga >> 32) & 0x01FFFFFFu)       // global_addr[56:32]
          | (2u << 30);                         // type = 2 ("image")
  i32x8 g1;
  g1[0] = (int)(2u << 16);                      // wg_mask=0, data_size=4B
  g1[1] = (int)((nFloats & 0xFFFFu) << 16);     // barrier_addr=0 | dim0.lo16
  g1[2] = (int)(((nFloats >> 16) & 0xFFFFu)     // dim0.hi16
          | (1u << 16));                        // tensor_dim1 = 1 (lo16)
  g1[3] = (int)((nFloats & 0xFFFFu) << 16);     // dim1.hi16=0 | tile_dim0
  g1[4] = 1;                                    // tile_dim1=1, tile_dim2=0
  g1[5] = (int)nFloats;                         // tensor_dim0_stride lo32
  g1[6] = 0;
  g1[7] = 0;
  i32x4 z4 = {0, 0, 0, 0};                      // groups 2/3 unused (<=2D)
  i32x8 z8 = {0, 0, 0, 0, 0, 0, 0, 0};
  __builtin_amdgcn_tensor_load_to_lds(g0, g1, z4, z4, z8, /*cpol=*/0);
}

// --- zero the (poisoned) output --------------------------------------------
__global__ void zero_out_kernel(float* out, int n) {
  int i = blockIdx.x * blockDim.x + threadIdx.x;
  if (i < n) out[i] = 0.0f;
}

// --- per-(batch, symmetry) affine transform: T[b][s] = 3x4 row-major -------
__global__ void precompute_transforms_kernel(const float* __restrict__ outp,
                                             float* __restrict__ T, int total) {
  int i = blockIdx.x * blockDim.x + threadIdx.x;
  if (i >= total) return;
  int s = i % 6;
  const float* o = outp + i * 4;
  float a0 = o[0], a1 = o[1], a2 = o[2], a3 = o[3];
  float M[12];
  if (s < 3) {
    // reflection: p - 2*((n.p + d)/|n|^2)*n
    float inv = 1.0f / (a0 * a0 + a1 * a1 + a2 * a2);
    float c = -2.0f * inv;
    M[0] = 1.0f + c * a0 * a0; M[1] = c * a0 * a1; M[2]  = c * a0 * a2; M[3]  = c * a3 * a0;
    M[4] = c * a1 * a0; M[5] = 1.0f + c * a1 * a1; M[6]  = c * a1 * a2; M[7]  = c * a3 * a1;
    M[8] = c * a2 * a0; M[9] = c * a2 * a1; M[10] = 1.0f + c * a2 * a2; M[11] = c * a3 * a2;
  } else {
    // rotation with _qinv = conj(q)/|q|  =>  A = R_raw(q)/|q|, t = 0
    float w = a0, x = a1, y = a2, z = a3;
    float k = rsqrtf(w * w + x * x + y * y + z * z);
    M[0]  = k * (w * w + x * x - y * y - z * z);
    M[1]  = k * 2.0f * (x * y - w * z);
    M[2]  = k * 2.0f * (x * z + w * y);
    M[3]  = 0.0f;
    M[4]  = k * 2.0f * (x * y + w * z);
    M[5]  = k * (w * w - x * x + y * y - z * z);
    M[6]  = k * 2.0f * (y * z - w * x);
    M[7]  = 0.0f;
    M[8]  = k * 2.0f * (x * z - w * y);
    M[9]  = k * 2.0f * (y * z + w * x);
    M[10] = k * (w * w - x * x - y * y + z * z);
    M[11] = 0.0f;
  }
#pragma unroll
  for (int r = 0; r < 12; ++r) T[i * 12 + r] = M[r];
}

// --- voxel gather + distance for one sym triple ----------------------------
__device__ __forceinline__ float sym_dist(float sx, float sy, float sz,
                                          const float* __restrict__ cbase) {
  float cx = fminf(fmaxf(sx, 0.0f), 32.0f);
  float cy = fminf(fmaxf(sy, 0.0f), 32.0f);
  float cz = fminf(fmaxf(sz, 0.0f), 32.0f);
  int ix = (int)cx; ix = ix > 31 ? 31 : ix;
  int iy = (int)cy; iy = iy > 31 ? 31 : iy;
  int iz = (int)cz; iz = iz > 31 ? 31 : iz;
  int flat = (ix << 10) + (iy << 5) + iz;
  const float* cp = cbase + flat * 3;          // L2-resident gather
  float dx = sx - cp[0];
  float dy = sy - cp[1];
  float dz = sz - cp[2];
  return sqrtf(fmaf(dx, dx, fmaf(dy, dy, dz * dz)));
}

// --- main ------------------------------------------------------------------
__global__ void __launch_bounds__(BLOCK_THREADS)
symmetry_loss_kernel(const float* __restrict__ points,
                     const float* __restrict__ closest,
                     const float* __restrict__ T, float* __restrict__ out,
                     int N, int G, float invB) {
  __shared__ float ldsPts[2][TILE_PTS * 3];   // 24 KB double buffer
  __shared__ float ldsT[72];                  // 6 transforms (3x4 each)
  __shared__ float ldsRed[BLOCK_THREADS / 32];

  const int b = blockIdx.y;
  const int tid = threadIdx.x;
  const int lane = tid & 31;
  const int waveId = tid >> 5;
  const int half = lane >> 4;                 // 0: lanes 0-15, 1: lanes 16-31
  const int row = lane & 15;

  if (tid < 72) ldsT[tid] = T[b * 72 + tid];

  const float* pbase = points +
      ((size_t)b * (size_t)N +
       (size_t)blockIdx.x * (TILE_PTS * TILES_PER_BLOCK)) * 3;
  const float* cbase = closest + (size_t)b * (size_t)G * 3;

  // Prologue: wave 0 DMAs tile 0 and waits for it.
  if (tid < 32) {
    tdm_load_tile(pbase, (u32)(uintptr_t)&ldsPts[0][0], TILE_PTS * 3);
    __builtin_amdgcn_s_wait_tensorcnt(0);
  }
  __syncthreads();   // ldsT + tile 0 visible

  // Build the stacked 18x4 transform as WMMA A-matrices (16x4 f32 layout:
  // lane L holds row L%16; VGPR0 = K0/K2, VGPR1 = K1/K3 for low/high half).
  v2f a0, a1;
  {
    int s = row / 3, c = row % 3;                  // rows 0..15
    const float* m = &ldsT[s * 12 + c * 4];
    a0.x = m[half ? 2 : 0];
    a0.y = m[half ? 3 : 1];
    int r2 = 16 + row;                             // rows 16..17 (t5.y, t5.z)
    if (r2 < 18) {
      int s2 = r2 / 3, c2 = r2 % 3;
      const float* m2 = &ldsT[s2 * 12 + c2 * 4];
      a1.x = m2[half ? 2 : 0];
      a1.y = m2[half ? 3 : 1];
    } else {
      a1.x = 0.0f; a1.y = 0.0f;
    }
  }

  float acc = 0.0f;
  for (int t = 0; t < TILES_PER_BLOCK; ++t) {
    // Kick off next tile's DMA into the other buffer (overlaps compute).
    if ((t + 1) < TILES_PER_BLOCK && tid < 32) {
      tdm_load_tile(pbase + (size_t)(t + 1) * TILE_PTS * 3,
                    (u32)(uintptr_t)&ldsPts[(t + 1) & 1][0], TILE_PTS * 3);
    }
    const float* buf = &ldsPts[t & 1][0];

#pragma unroll
    for (int g = 0; g < PTS_PER_WAVE / 16; ++g) {
      const int p = waveId * PTS_PER_WAVE + g * 16 + row;  // this lane's point
      // B-matrix (4x16 f32): low half supplies (x,y)=K0,K1 of its point,
      // high half supplies (z,1)=K2,K3.
      v2f bv;
      int bi = p * 3 + (half ? 2 : 0);
      bv.x = buf[bi];
      bv.y = half ? 1.0f : buf[bi + 1];

      v8f cz = {0.f, 0.f, 0.f, 0.f, 0.f, 0.f, 0.f, 0.f};
      // D(16x16): lane = point column L%16; low half rows 0-7, high rows 8-15.
      v8f d0 = __builtin_amdgcn_wmma_f32_16x16x4_f32(
          false, a0, false, bv, (short)0, cz, false, false);
      v8f d1 = __builtin_amdgcn_wmma_f32_16x16x4_f32(
          false, a1, false, bv, (short)0, cz, false, false);

      // Redistribute split triples across the half-wave boundary.
      float x_t2z = __shfl_xor(d0[0], 16, 32);  // low gets high's row8 = t2.z
      float x_t5y = __shfl_xor(d1[0], 16, 32);  // high gets low's row16 = t5.y
      float x_t5z = __shfl_xor(d1[1], 16, 32);  // high gets low's row17 = t5.z

      float sxa, sya, sza, sxb, syb, szb, sxc, syc, szc;
      if (half == 0) {          // rows 0-7: t0, t1, t2.xy (+t2.z via shfl)
        sxa = d0[0]; sya = d0[1]; sza = d0[2];
        sxb = d0[3]; syb = d0[4]; szb = d0[5];
        sxc = d0[6]; syc = d0[7]; szc = x_t2z;
      } else {                  // rows 8-15: t3, t4, t5.x (+t5.yz via shfl)
        sxa = d0[1]; sya = d0[2]; sza = d0[3];
        sxb = d0[4]; syb = d0[5]; szb = d0[6];
        sxc = d0[7]; syc = x_t5y; szc = x_t5z;
      }
      acc += sym_dist(sxa, sya, sza, cbase);
      acc += sym_dist(sxb, syb, szb, cbase);
      acc += sym_dist(sxc, syc, szc, cbase);
    }
    if (tid < 32) __builtin_amdgcn_s_wait_tensorcnt(0);
    __syncthreads();
  }

  // wave32 shuffle reduction, then cross-wave via LDS
#pragma unroll
  for (int off = 16; off > 0; off >>= 1) acc += __shfl_down(acc, off, 32);
  if ((tid & 31) == 0) ldsRed[tid >> 5] = acc;
  __syncthreads();
  if (tid < BLOCK_THREADS / 32) {
    float v = ldsRed[tid];
#pragma unroll
    for (int off = (BLOCK_THREADS / 64); off > 0; off >>= 1)
      v += __shfl_down(v, off, BLOCK_THREADS / 32);
    if (tid == 0) atomicAdd(out, v * invB);
  }
}

extern "C" void kernel_launch(void* const* d_in, const int* in_sizes, int n_in,
                              void* d_out, int out_size, void* d_ws,
                              size_t ws_size, hipStream_t stream) {
  const float* outp    = (const float*)d_in[0];   // (B, 6, 4)
  const float* points  = (const float*)d_in[1];   // (B, N, 3)
  const float* closest = (const float*)d_in[2];   // (B, G, 3)
  float* out = (float*)d_out;

  const int B = in_sizes[0] / 24;                 // 16
  const int N = (int)((long long)in_sizes[1] / (3LL * B));   // 131072
  const int G = (int)((long long)in_sizes[2] / (3LL * B));   // 32768

  float* T = (float*)d_ws;                        // B*6*12 floats (4.5 KB)

  zero_out_kernel<<<1, 64, 0, stream>>>(out, out_size);
  precompute_transforms_kernel<<<1, 128, 0, stream>>>(outp, T, B * 6);

  const int ptsPerBlock = TILE_PTS * TILES_PER_BLOCK;  // 4096, divides N
  dim3 grid(N / ptsPerBlock, B);                       // (32, 16)
  symmetry_loss_kernel<<<grid, BLOCK_THREADS, 0, stream>>>(
      points, closest, T, out, N, G, 1.0f / (float)B);
}